// IGMC_9242769621423
// MI455X (gfx1250) — compile-verified
//
#include <hip/hip_runtime.h>
#include <hip/hip_bf16.h>

typedef __attribute__((ext_vector_type(2))) float v2f;
typedef __attribute__((ext_vector_type(8))) float v8f;

#define N_NODES 100000
#define N_EDGES 3200000
#define N_REL 5

// ---------------------------------------------------------------------------
// 1) Per-(dst,rel) edge counts via float atomics, then reciprocal in place.
// ---------------------------------------------------------------------------
__global__ void count_kernel(const int* __restrict__ dst, const int* __restrict__ et,
                             float* __restrict__ cnt, int nE) {
    int e = blockIdx.x * blockDim.x + threadIdx.x;
    if (e < nE) atomicAdd(&cnt[dst[e] * N_REL + et[e]], 1.0f);
}

__global__ void inv_kernel(float* __restrict__ cnt, int n) {
    int i = blockIdx.x * blockDim.x + threadIdx.x;
    if (i < n) { float c = cnt[i]; cnt[i] = 1.0f / fmaxf(c, 1.0f); }
}

// ---------------------------------------------------------------------------
// 2) Combined weight: rows [0,5*IN) = sum_b comp[r,b]*basis[b,i,:] (row r*IN+i)
//    rows [5*IN,6*IN) = root[i,:]. Shape [6*IN, 32].
// ---------------------------------------------------------------------------
__global__ void wcat_kernel(const float* __restrict__ basis, const float* __restrict__ comp,
                            const float* __restrict__ root, float* __restrict__ wcat, int IN) {
    int idx = blockIdx.x * blockDim.x + threadIdx.x;
    int K = 6 * IN;
    if (idx >= K * 32) return;
    int row = idx >> 5;
    int o   = idx & 31;
    if (row < 5 * IN) {
        int r = row / IN, i = row % IN;
        float acc = 0.0f;
#pragma unroll
        for (int b = 0; b < 4; ++b)
            acc += comp[r * 4 + b] * basis[(b * IN + i) * 32 + o];
        wcat[idx] = acc;
    } else {
        wcat[idx] = root[(row - 5 * IN) * 32 + o];
    }
}

// ---------------------------------------------------------------------------
// 3) Edge scatter: one thread per (edge, channel). Lanes of a wave cover
//    consecutive channels -> coalesced gather of x[src] row and coalesced
//    global_atomic_add_f32 into agg[(dst*5+rel)*IN + c].
// ---------------------------------------------------------------------------
template <int IN>
__global__ void scatter_kernel(const int* __restrict__ src, const int* __restrict__ dst,
                               const int* __restrict__ et, const float* __restrict__ xin,
                               int ldx, float* __restrict__ agg, int nE) {
    long long idx = (long long)blockIdx.x * blockDim.x + threadIdx.x;
    if (idx >= (long long)nE * IN) return;
    int e = (int)(idx / IN);
    int c = (int)(idx % IN);
    int s = src[e], d = dst[e], r = et[e];
    atomicAdd(&agg[(d * N_REL + r) * IN + c], xin[s * ldx + c]);
}

// ---------------------------------------------------------------------------
// 4) Node transform via V_WMMA_F32_16X16X4_F32.
//    One wave (32 threads) per 16-node tile. A row for node n:
//      [ invcnt[n,r]*agg[n,r,i] for r,i ] ++ [ x[n,i] ]   (length K = 6*IN)
//    Output 16x32 = two 16x16 f32 tiles; + bias, tanh, store into h[:,col0..].
//    A frag (16x4 f32): lane m=tid&15; lanes 16-31 carry K+2/K+3.
//    B frag (4x16 f32): symmetric (lane = column, VGPRs = K rows).
// ---------------------------------------------------------------------------
template <int IN>
__global__ void __launch_bounds__(32)
transform_kernel(const float* __restrict__ agg, const float* __restrict__ inv,
                 const float* __restrict__ xin, int ldx,
                 const float* __restrict__ wcat, const float* __restrict__ bias,
                 float* __restrict__ h, int col0) {
    constexpr int K  = 6 * IN;
    constexpr int KP = K + 1;                 // pad to avoid LDS bank conflicts
    __shared__ float Als[16 * KP];
    __shared__ float Wls[K * 32];

    const int tid   = threadIdx.x;
    const int node0 = blockIdx.x * 16;

    // stage combined weight [K,32] into LDS
    for (int j = tid; j < K * 32; j += 32) Wls[j] = wcat[j];

    // stage A rows into LDS (invcnt-scaled agg ++ x)
    for (int m = 0; m < 16; ++m) {
        int n = node0 + m;
        const float* aggn = agg + (long long)n * N_REL * IN;
        const float* invn = inv + n * N_REL;
        for (int k = tid; k < K; k += 32) {
            float v;
            if (k < 5 * IN) {
                int r = k / IN, i = k % IN;   // IN is constexpr -> shifts
                v = invn[r] * aggn[r * IN + i];
            } else {
                v = xin[(long long)n * ldx + (k - 5 * IN)];
            }
            Als[m * KP + k] = v;
        }
    }
    __syncthreads();

    const int mlane = tid & 15;
    const int koff  = (tid >> 4) * 2;         // lanes 16-31 hold K+2, K+3
    v8f acc0 = {0.f, 0.f, 0.f, 0.f, 0.f, 0.f, 0.f, 0.f};
    v8f acc1 = {0.f, 0.f, 0.f, 0.f, 0.f, 0.f, 0.f, 0.f};

    for (int k0 = 0; k0 < K; k0 += 4) {
        v2f a;
        a.x = Als[mlane * KP + k0 + koff];
        a.y = Als[mlane * KP + k0 + koff + 1];
        v2f b0, b1;
        b0.x = Wls[(k0 + koff) * 32 + mlane];
        b0.y = Wls[(k0 + koff + 1) * 32 + mlane];
        b1.x = Wls[(k0 + koff) * 32 + 16 + mlane];
        b1.y = Wls[(k0 + koff + 1) * 32 + 16 + mlane];
        acc0 = __builtin_amdgcn_wmma_f32_16x16x4_f32(false, a, false, b0,
                                                     (short)0, acc0, false, false);
        acc1 = __builtin_amdgcn_wmma_f32_16x16x4_f32(false, a, false, b1,
                                                     (short)0, acc1, false, false);
    }

    // D layout: lanes 0-15 -> M = v, lanes 16-31 -> M = v + 8; N = lane & 15
    const int mbase = (tid >> 4) * 8;
#pragma unroll
    for (int v = 0; v < 8; ++v) {
        int m = mbase + v;
        long long off = (long long)(node0 + m) * 128 + col0;
        h[off + mlane]      = tanhf(acc0[v] + bias[mlane]);
        h[off + 16 + mlane] = tanhf(acc1[v] + bias[16 + mlane]);
    }
}

// ---------------------------------------------------------------------------
// 5) Readout MLP: one block (128 threads) per graph.
//    g = [h[u] ++ h[v]] (256), z = relu(g@w1 + b1), out = z@w2 + b2.
// ---------------------------------------------------------------------------
__global__ void __launch_bounds__(128)
mlp_kernel(const float* __restrict__ h, const int* __restrict__ uidx,
           const int* __restrict__ iidx, const float* __restrict__ w1,
           const float* __restrict__ b1, const float* __restrict__ w2,
           const float* __restrict__ b2, float* __restrict__ out) {
    __shared__ float g[256];
    __shared__ float red[128];
    int b   = blockIdx.x;
    int tid = threadIdx.x;
    int u = uidx[b], it = iidx[b];
    g[tid]       = h[(long long)u * 128 + tid];
    g[128 + tid] = h[(long long)it * 128 + tid];
    __syncthreads();

    float acc = b1[tid];
#pragma unroll 8
    for (int k = 0; k < 256; ++k) acc += g[k] * w1[k * 128 + tid];
    float z = acc > 0.0f ? acc : 0.0f;

    red[tid] = z * w2[tid];
    __syncthreads();
    for (int s = 64; s > 0; s >>= 1) {
        if (tid < s) red[tid] += red[tid + s];
        __syncthreads();
    }
    if (tid == 0) out[b] = red[0] + b2[0];
}

// ---------------------------------------------------------------------------
extern "C" void kernel_launch(void* const* d_in, const int* in_sizes, int n_in,
                              void* d_out, int out_size, void* d_ws, size_t ws_size,
                              hipStream_t stream) {
    (void)in_sizes; (void)n_in; (void)out_size; (void)ws_size;

    const float* x     = (const float*)d_in[0];
    const int*   eidx  = (const int*)d_in[1];
    const int*   etype = (const int*)d_in[2];
    const int*   uidx  = (const int*)d_in[3];
    const int*   iidx  = (const int*)d_in[4];
    const float* basis[4] = {(const float*)d_in[5],  (const float*)d_in[9],
                             (const float*)d_in[13], (const float*)d_in[17]};
    const float* comp[4]  = {(const float*)d_in[6],  (const float*)d_in[10],
                             (const float*)d_in[14], (const float*)d_in[18]};
    const float* root[4]  = {(const float*)d_in[7],  (const float*)d_in[11],
                             (const float*)d_in[15], (const float*)d_in[19]};
    const float* bias[4]  = {(const float*)d_in[8],  (const float*)d_in[12],
                             (const float*)d_in[16], (const float*)d_in[20]};
    const float* w1 = (const float*)d_in[21];
    const float* b1 = (const float*)d_in[22];
    const float* w2 = (const float*)d_in[23];
    const float* b2 = (const float*)d_in[24];
    float* out = (float*)d_out;

    const int N = N_NODES, E = N_EDGES;
    const int* src = eidx;
    const int* dst = eidx + E;

    // workspace layout (floats): inv[N*5] | agg[N*5*32] | h[N*128] | wcat[6144]
    float* inv = (float*)d_ws;
    float* agg = inv + (size_t)N * N_REL;
    float* h   = agg + (size_t)N * N_REL * 32;
    float* wc  = h   + (size_t)N * 128;

    hipMemsetAsync(inv, 0, (size_t)N * N_REL * sizeof(float), stream);
    count_kernel<<<(E + 255) / 256, 256, 0, stream>>>(dst, etype, inv, E);
    inv_kernel<<<(N * N_REL + 255) / 256, 256, 0, stream>>>(inv, N * N_REL);

    for (int l = 0; l < 4; ++l) {
        const int IN = (l == 0) ? 4 : 32;
        const int K  = 6 * IN;
        const float* xin = (l == 0) ? x : (h + (size_t)(l - 1) * 32);
        const int    ldx = (l == 0) ? 4 : 128;

        hipMemsetAsync(agg, 0, (size_t)N * N_REL * IN * sizeof(float), stream);
        wcat_kernel<<<(K * 32 + 255) / 256, 256, 0, stream>>>(basis[l], comp[l], root[l], wc, IN);

        int sblocks = (int)(((long long)E * IN + 255) / 256);
        if (IN == 4) {
            scatter_kernel<4><<<sblocks, 256, 0, stream>>>(src, dst, etype, xin, ldx, agg, E);
            transform_kernel<4><<<N / 16, 32, 0, stream>>>(agg, inv, xin, ldx, wc, bias[l], h, l * 32);
        } else {
            scatter_kernel<32><<<sblocks, 256, 0, stream>>>(src, dst, etype, xin, ldx, agg, E);
            transform_kernel<32><<<N / 16, 32, 0, stream>>>(agg, inv, xin, ldx, wc, bias[l], h, l * 32);
        }
    }

    mlp_kernel<<<4096, 128, 0, stream>>>(h, uidx, iidx, w1, b1, w2, b2, out);
}